// DecodePredictions_45938970198445
// MI455X (gfx1250) — compile-verified
//
#include <hip/hip_runtime.h>
#include <cstdint>

#define NANCH 76725
#define NCLS  80
#define KCAND 512
#define BLK   512

typedef float v2f __attribute__((ext_vector_type(2)));
typedef float v8f __attribute__((ext_vector_type(8)));

// order-preserving float -> uint map (descending top-k on uint)
__device__ __forceinline__ unsigned okey(float x) {
  unsigned b = __float_as_uint(x);
  return (b & 0x80000000u) ? ~b : (b | 0x80000000u);
}
__device__ __forceinline__ float okey_inv(unsigned u) {
  unsigned b = (u & 0x80000000u) ? (u ^ 0x80000000u) : ~u;
  return __uint_as_float(b);
}
__device__ __forceinline__ unsigned ballot32(bool p) {
  return __builtin_amdgcn_ballot_w32(p);
}

// ---------------------------------------------------------------------------
// Kernel 1: per (batch, class) top-512 radix select -> sort -> decode -> NMS
// grid = (80, 4), block = 512 threads (16 waves)
// ---------------------------------------------------------------------------
__global__ __launch_bounds__(BLK) void topk_nms_kernel(
    const float* __restrict__ pred, float* __restrict__ wsS, float* __restrict__ wsB) {
  const int c = blockIdx.x;
  const int b = blockIdx.y;
  const int tid = threadIdx.x;
  const float* col = pred + (size_t)b * NANCH * 84 + c;  // element a at col[a*84]

  __shared__ unsigned hist[256];
  __shared__ unsigned sSel, sNeed, sCntGt, sCntEq;
  __shared__ unsigned long long skey[KCAND];   // (okey<<32) | ~anchor_idx
  __shared__ float4 sbox[KCAND];
  __shared__ float  sscore[KCAND];
  __shared__ float  sarea[KCAND];
  __shared__ unsigned bitmap[KCAND * 16];      // 512 rows x 512 suppression bits
  __shared__ unsigned keepmask[16];
  __shared__ unsigned wpfx[17];

  if (tid == 0) { sNeed = KCAND; sCntGt = 0; sCntEq = 0; }

  // ---- 4-pass MSB radix select: exact key T of the 512th largest logit ----
  unsigned prefix = 0, maskhi = 0;
  for (int pass = 0; pass < 4; ++pass) {
    const int shift = 24 - 8 * pass;
    for (int i = tid; i < 256; i += BLK) hist[i] = 0;
    __syncthreads();
    for (int a = tid; a < NANCH; a += BLK) {
      unsigned u = okey(col[(size_t)a * 84]);
      if ((u & maskhi) == prefix) atomicAdd(&hist[(u >> shift) & 0xFFu], 1u);
    }
    __syncthreads();
    if (tid == 0) {
      unsigned need = sNeed, cum = 0, sel = 0;
      for (int bin = 255; bin >= 0; --bin) {
        unsigned cn = hist[bin];
        if (cum + cn >= need) { sel = (unsigned)bin; sNeed = need - cum; break; }
        cum += cn;
      }
      sSel = sel;
    }
    __syncthreads();
    prefix |= sSel << shift;
    maskhi |= 0xFFu << shift;
    __syncthreads();
  }
  const unsigned T    = prefix;
  const unsigned need = sNeed;         // #elements to take from the ==T group
  const unsigned ngt  = KCAND - need;  // #elements strictly above T

  // ---- compaction into 512 slots ----
  for (int a = tid; a < NANCH; a += BLK) {
    unsigned u = okey(col[(size_t)a * 84]);
    if (u > T) {
      unsigned p = atomicAdd(&sCntGt, 1u);
      skey[p] = ((unsigned long long)u << 32) | (unsigned)(~(unsigned)a);
    } else if (u == T) {
      unsigned r = atomicAdd(&sCntEq, 1u);
      if (r < need)
        skey[ngt + r] = ((unsigned long long)u << 32) | (unsigned)(~(unsigned)a);
    }
  }
  __syncthreads();

  // ---- bitonic sort, descending (ties: lowest anchor index first via ~idx) ----
  for (unsigned k = 2; k <= KCAND; k <<= 1) {
    for (unsigned j = k >> 1; j > 0; j >>= 1) {
      unsigned i = (unsigned)tid, p = i ^ j;
      if (p > i) {
        unsigned long long x = skey[i], y = skey[p];
        bool up = ((i & k) == 0);
        if (up ? (x < y) : (x > y)) { skey[i] = y; skey[p] = x; }
      }
      __syncthreads();
    }
  }

  // ---- decode boxes + sigmoid for the 512 survivors only ----
  {
    unsigned long long key = skey[tid];
    unsigned u   = (unsigned)(key >> 32);
    unsigned idx = ~((unsigned)key);
    float logit  = okey_inv(u);
    float score  = 1.0f / (1.0f + __expf(-logit));

    int base, fw, stride;
    if      (idx < 57600u) { base = 0;     fw = 80; stride = 8;   }
    else if (idx < 72000u) { base = 57600; fw = 40; stride = 16;  }
    else if (idx < 75600u) { base = 72000; fw = 20; stride = 32;  }
    else if (idx < 76500u) { base = 75600; fw = 10; stride = 64;  }
    else                   { base = 76500; fw = 5;  stride = 128; }
    int r    = (int)idx - base;
    int k9   = r % 9, cell = r / 9;
    int ix   = cell % fw, iy = cell / fw;
    float cx = (ix + 0.5f) * (float)stride;
    float cy = (iy + 0.5f) * (float)stride;
    const float sqr[3] = {0.70710678f, 1.0f, 1.41421356f};        // sqrt(ratio)
    const float scl[3] = {1.0f, 1.25992105f, 1.58740105f};        // 2^(s/3)
    float s4 = 4.0f * (float)stride * scl[k9 % 3];
    float aw = s4 * sqr[k9 / 3];
    float ah = s4 / sqr[k9 / 3];

    const float4 q = *(const float4*)(pred + ((size_t)b * NANCH + idx) * 84 + 80);
    float bx = q.x * 0.1f * aw + cx;
    float by = q.y * 0.1f * ah + cy;
    float bw = __expf(q.z * 0.2f) * aw;
    float bh = __expf(q.w * 0.2f) * ah;
    sbox[tid]   = make_float4(bx, by, bw, bh);
    sscore[tid] = score;
    // reference's IoU treats [cx,cy,w,h] as [y1,x1,y2,x2]; replicate exactly
    sarea[tid]  = fmaxf(bw - bx, 0.f) * fmaxf(bh - by, 0.f);
    unsigned m0 = ballot32(score > 0.05f);
    if ((tid & 31) == 0) keepmask[tid >> 5] = m0;
  }
  for (int i = tid; i < KCAND * 16; i += BLK) bitmap[i] = 0;
  __syncthreads();

  // ---- suppression bitmap: union base via WMMA outer-sum (area_i + area_j) ----
  {
    const int wave  = tid >> 5;
    const int lane  = tid & 31;
    const int laneLo = lane & 15;
    const bool hi   = lane >= 16;
    for (int t = wave; t < 1024; t += 16) {             // 32x32 grid of 16x16 tiles
      const int rowbase = (t >> 5) << 4;
      const int colbase = (t & 31) << 4;
      v2f A, Bv;
      A.x  = hi ? 0.f : sarea[rowbase + laneLo];        // A: 16x4, K0 = area_row
      A.y  = hi ? 0.f : 1.0f;                           //         K1 = 1
      Bv.x = hi ? 0.f : 1.0f;                           // B: 4x16, K0 = 1
      Bv.y = hi ? 0.f : sarea[colbase + laneLo];        //         K1 = area_col
      v8f C = {};
      v8f D = __builtin_amdgcn_wmma_f32_16x16x4_f32(false, A, false, Bv,
                                                    (short)0, C, false, false);
      const int N = colbase + laneLo;
      const float4 bj = sbox[N];
#pragma unroll
      for (int rr = 0; rr < 8; ++rr) {
        const int M = rowbase + rr + (hi ? 8 : 0);      // documented C/D lane layout
        const float4 bi = sbox[M];
        float ih    = fmaxf(fminf(bi.z, bj.z) - fmaxf(bi.x, bj.x), 0.f);
        float iw    = fmaxf(fminf(bi.w, bj.w) - fmaxf(bi.y, bj.y), 0.f);
        float inter = ih * iw;
        float uni   = D[rr] - inter;                    // area_i + area_j - inter
        bool  sup   = (uni > 0.f) && (inter > 0.5f * uni) && (N > M);
        unsigned m  = ballot32(sup);
        if (lane == 0)
          atomicOr(&bitmap[(rowbase + rr) * 16 + (colbase >> 5)],
                   (m & 0xFFFFu) << (colbase & 16));
        else if (lane == 16)
          atomicOr(&bitmap[(rowbase + 8 + rr) * 16 + (colbase >> 5)],
                   (m >> 16) << (colbase & 16));
      }
    }
  }
  __syncthreads();

  // ---- sequential greedy reduction over the bitmap (wave 0, DS-ordered) ----
  if (tid < 32) {
    volatile unsigned* km = keepmask;
    for (int i = 0; i < KCAND; ++i) {
      unsigned kw = km[i >> 5];
      if ((kw >> (i & 31)) & 1u) {
        if (tid < 16) km[tid] &= ~bitmap[i * 16 + tid];
      }
    }
  }
  __syncthreads();
  if (tid == 0) {
    unsigned s = 0;
    for (int w = 0; w < 16; ++w) { wpfx[w] = s; s += __popc(keepmask[w]); }
    wpfx[16] = s;
  }
  __syncthreads();

  // ---- compact kept (sorted order) into per-class top-100 workspace ----
  {
    unsigned kw   = keepmask[tid >> 5];
    bool keep     = (kw >> (tid & 31)) & 1u;
    unsigned rank = wpfx[tid >> 5] + __popc(kw & ((1u << (tid & 31)) - 1u));
    float*  oS = wsS + (size_t)(b * NCLS + c) * 100;
    float4* oB = (float4*)wsB + (size_t)(b * NCLS + c) * 100;
    if (keep && rank < 100) { oS[rank] = sscore[tid]; oB[rank] = sbox[tid]; }
    unsigned kept = wpfx[16] < 100u ? wpfx[16] : 100u;
    if (tid < 100 && (unsigned)tid >= kept) {
      oS[tid] = 0.f;
      oB[tid] = make_float4(0.f, 0.f, 0.f, 0.f);
    }
  }
}

// ---------------------------------------------------------------------------
// Kernel 2: per batch, global top-100 over 80x100 class results
// grid = 4, block = 1024. Scores staged via async tensor path to LDS.
// ---------------------------------------------------------------------------
__global__ __launch_bounds__(1024) void combine_kernel(
    const float* __restrict__ wsS, const float* __restrict__ wsB, float* __restrict__ out) {
  const int b   = blockIdx.x;
  const int tid = threadIdx.x;
  __shared__ unsigned long long keys[8192];      // 64 KB
  float* sS = (float*)&keys[4096];               // 32 KB staging in upper half
  const float* gS = wsS + (size_t)b * 8000;

  // async copy 8000 floats (2000 x 16B chunks) global -> LDS
  unsigned ldsBase = (unsigned)(uintptr_t)sS;
  for (int ch = tid; ch < 2000; ch += 1024) {
    unsigned ldsAddr = ldsBase + (unsigned)ch * 16u;
    unsigned gOff    = (unsigned)ch * 16u;
    asm volatile("global_load_async_to_lds_b128 %0, %1, %2"
                 :: "v"(ldsAddr), "v"(gOff), "s"(gS) : "memory");
  }
  asm volatile("s_wait_asynccnt 0x0" ::: "memory");
  __syncthreads();

  float regs[8];
#pragma unroll
  for (int p = 0; p < 8; ++p) {
    int i = tid + (p << 10);
    regs[p] = (i < 8000) ? sS[i] : 0.f;
  }
  __syncthreads();
#pragma unroll
  for (int p = 0; p < 8; ++p) {
    int i = tid + (p << 10);
    if (i < 8000)
      keys[i] = ((unsigned long long)__float_as_uint(regs[p]) << 32) |
                (unsigned)(~(unsigned)i);   // scores >= 0: raw bits are ordered
    else
      keys[i] = 0ull;
  }
  __syncthreads();

  // bitonic sort 8192, descending
  for (unsigned k = 2; k <= 8192; k <<= 1) {
    for (unsigned j = k >> 1; j > 0; j >>= 1) {
#pragma unroll
      for (int p = 0; p < 8; ++p) {
        unsigned i = (unsigned)tid + ((unsigned)p << 10);
        unsigned q = i ^ j;
        if (q > i) {
          unsigned long long x = keys[i], y = keys[q];
          bool up = ((i & k) == 0);
          if (up ? (x < y) : (x > y)) { keys[i] = y; keys[q] = x; }
        }
      }
      __syncthreads();
    }
  }

  // emit: boxes[4,100,4] | scores[4,100] | classes[4,100] | valid[4]
  if (tid < 100) {
    unsigned long long key = keys[tid];
    float s = __uint_as_float((unsigned)(key >> 32));
    bool valid = s > 0.f;
    float bx = 0.f, by = 0.f, bw = 0.f, bh = 0.f, cls = 0.f;
    if (valid) {
      unsigned flat = ~((unsigned)key);                 // c*100 + slot, < 8000
      const float* bb = wsB + ((size_t)b * 8000 + flat) * 4;
      bx = fminf(fmaxf(bb[0], 0.f), 1.f);
      by = fminf(fmaxf(bb[1], 0.f), 1.f);
      bw = fminf(fmaxf(bb[2], 0.f), 1.f);
      bh = fminf(fmaxf(bb[3], 0.f), 1.f);
      cls = (float)(flat / 100u);
    }
    float* ob = out + (size_t)(b * 100 + tid) * 4;
    ob[0] = bx; ob[1] = by; ob[2] = bw; ob[3] = bh;
    out[1600 + b * 100 + tid] = valid ? s : 0.f;
    out[2000 + b * 100 + tid] = cls;
    // valid count: sorted descending => positives form a prefix
    bool nxt = (tid == 99) ? false
                           : (__uint_as_float((unsigned)(keys[tid + 1] >> 32)) > 0.f);
    if (valid && !nxt) out[2400 + b] = (float)(tid + 1);
    if (tid == 0 && !valid) out[2400 + b] = 0.f;
  }
}

extern "C" void kernel_launch(void* const* d_in, const int* in_sizes, int n_in,
                              void* d_out, int out_size, void* d_ws, size_t ws_size,
                              hipStream_t stream) {
  (void)in_sizes; (void)n_in; (void)out_size; (void)ws_size;
  const float* pred = (const float*)d_in[1];       // d_in[0] = images (shape only)
  float* wsS = (float*)d_ws;                       // 4*80*100 scores   (128 KB)
  float* wsB = (float*)d_ws + 32000;               // 4*80*100*4 boxes  (512 KB)
  dim3 g(NCLS, 4);
  topk_nms_kernel<<<g, BLK, 0, stream>>>(pred, wsS, wsB);
  combine_kernel<<<4, 1024, 0, stream>>>(wsS, wsB, (float*)d_out);
}